// SignatureToLogSignature_73830487818380
// MI455X (gfx1250) — compile-verified
//
#include <hip/hip_runtime.h>
#include <cstdint>

// ---------------------------------------------------------------------------
// Log-signature (Lyndon 'words' mode), C=8 channels, depth 6, batch 128.
//
// log(1+x) = x + x (.) Q,  Q = sum_{n=1..5} (-1)^n/(n+1) x^n (truncated).
// Q built per batch element in LDS via Horner (4 truncated tensor products);
// each Lyndon output = x_k[j] + sum_i x_i[prefix_i] * Q_{k-i}[suffix_i].
// LDS/block: x levels 1..4 (4680 f32) + Q levels 1..5 (37448 f32) = 168.5 KB.
// The 64x512 and 512x64 outer-product terms of the level-5 product run on
// V_WMMA_F32_16X16X4_F32 (K=1 outer product, zero-padded K slices, C-operand
// accumulation).
// ---------------------------------------------------------------------------

#define NT 1024

typedef float v2f __attribute__((ext_vector_type(2)));
typedef float v8f __attribute__((ext_vector_type(8)));

// offset of level k inside the "levels 1..5" concatenation: (8^k - 8)/7
constexpr int LVL_OFF(int k) { return ((1 << (3 * k)) - 8) / 7; }

// Lyndon output ranges per level (counts 8/28/168/1008/6552/43596)
constexpr int OUT_BEG(int k) {
  return (k == 1) ? 0 : (k == 2) ? 8 : (k == 3) ? 36 : (k == 4) ? 204
                                                     : (k == 5) ? 1212 : 7764;
}
constexpr int OUT_END(int k) { return (k == 6) ? 51360 : OUT_BEG(k + 1); }

// ---------------------------------------------------------------------------
// Kernel 1: Lyndon index table (deterministic ordered ballot compaction).
// table[pos] = (k << 20) | j, length-major, lex within length.
// ---------------------------------------------------------------------------
__global__ void __launch_bounds__(NT) lyndon_kernel(unsigned* __restrict__ lyn) {
  __shared__ unsigned wsum[32];
  __shared__ unsigned woff[32];
  __shared__ unsigned base, tot;
  const int k = (int)blockIdx.x + 1;
  const unsigned N = 1u << (3 * k);
  const unsigned outoffs[6] = {0u, 8u, 36u, 204u, 1212u, 7764u};
  const unsigned outoff = outoffs[blockIdx.x];
  const int tid = (int)threadIdx.x;
  const int lane = tid & 31;
  const int wid = tid >> 5;
  if (tid == 0) base = 0u;
  __syncthreads();
  for (unsigned start = 0; start < N; start += (unsigned)NT) {
    unsigned j = start + (unsigned)tid;
    int flag = 0;
    if (j < N) {
      flag = 1;
      for (int r = 1; r < k; ++r) {
        int s = 3 * (k - r);
        unsigned rot = ((j & ((1u << s) - 1u)) << (3 * r)) | (j >> s);
        if (rot <= j) { flag = 0; break; }
      }
    }
    unsigned mask = (unsigned)__ballot(flag);  // wave32: low 32 bits
    unsigned prefix = (unsigned)__popc(mask & ((1u << lane) - 1u));
    if (lane == 0) wsum[wid] = (unsigned)__popc(mask);
    __syncthreads();
    if (tid == 0) {
      unsigned acc = 0;
      for (int w = 0; w < 32; ++w) { woff[w] = acc; acc += wsum[w]; }
      tot = acc;
    }
    __syncthreads();
    if (flag) lyn[outoff + base + woff[wid] + prefix] = ((unsigned)k << 20) | j;
    __syncthreads();
    if (tid == 0) base += tot;
    __syncthreads();
  }
}

// ---------------------------------------------------------------------------
// WMMA outer-product accumulate: dst[R x Cc] += p[R] (outer) q[Cc].
// 16x16 tiles, K=1 via V_WMMA_F32_16X16X4_F32; tile accumulated through the
// C operand. Constexpr trip count -> straight-line, EXEC stays all-ones.
// ---------------------------------------------------------------------------
template <int R, int Cc>
__device__ __forceinline__ void wmma_outer_acc(const float* __restrict__ p,
                                               const float* __restrict__ q,
                                               float* __restrict__ dst, int tid) {
  const int lane = tid & 31;
  const int wid = tid >> 5;
  const int ln = lane & 15;
  const float msk = (lane < 16) ? 1.0f : 0.0f;  // zero K!=0 operand slices
  constexpr int ntc = Cc >> 4;
  constexpr int iters = ((R >> 4) * ntc) / 32;  // 128 tiles / 32 waves = 4
#pragma unroll
  for (int it = 0; it < iters; ++it) {
    int t = wid + it * 32;                      // uniform per wave
    int tr = t / ntc;
    int tc = t & (ntc - 1);
    // A (16x4): VGPR0 lanes0-15 = K=0 column (M=lane), rest zero.
    v2f a;  a.x = p[tr * 16 + ln] * msk;  a.y = 0.0f;
    // B (4x16): VGPR0 lanes0-15 = K=0 row (N=lane), rest zero.
    v2f bv; bv.x = q[tc * 16 + ln] * msk; bv.y = 0.0f;
    // C/D layout: VGPR r -> M=r (lanes0-15) / M=r+8 (lanes16-31), N=lane&15
    int M0 = tr * 16 + ((lane >> 4) << 3);
    int Ncol = tc * 16 + ln;
    v8f c;
#pragma unroll
    for (int r = 0; r < 8; ++r) c[r] = dst[(M0 + r) * Cc + Ncol];
    v8f d = __builtin_amdgcn_wmma_f32_16x16x4_f32(false, a, false, bv,
                                                  (short)0, c, false, false);
#pragma unroll
    for (int r = 0; r < 8; ++r) dst[(M0 + r) * Cc + Ncol] = d[r];
  }
}

// level-K slot of U <- (a + x (.) U), K <= 4, fully unrolled inner terms:
// u_K[m] = U0*x_K[m] + sum_{i=1..K-1} x_i[m >> 3(K-i)] * u_{K-i}[m & mask]
template <int K>
__device__ __forceinline__ void mul_level(const float* __restrict__ xl14,
                                          float* __restrict__ u, float U0, int tid) {
  constexpr int n = 1 << (3 * K);
  constexpr int xo = LVL_OFF(K);
  for (int m = tid; m < n; m += NT) {
    float acc = U0 * xl14[xo + m];
#pragma unroll
    for (int i = 1; i < K; ++i) {
      const int s = 3 * (K - i);
      acc += xl14[LVL_OFF(i) + (m >> s)] * u[LVL_OFF(K - i) + (m & ((1 << s) - 1))];
    }
    u[xo + m] = acc;  // safe in place: only u levels < K are read
  }
}

// level-5 slot (final Horner step only): scalar i=1,i=4 + WMMA i=2,i=3
__device__ __forceinline__ void mul_level5(const float* __restrict__ xl14,
                                           float* __restrict__ u,
                                           const float* __restrict__ x5g,
                                           float U0, int tid) {
  for (int m = tid; m < 32768; m += NT) {
    float acc = U0 * x5g[m];                     // scalar term, coalesced global
    acc += xl14[m >> 12] * u[584 + (m & 4095)];  // i=1: x1 (outer) u4
    acc += xl14[584 + (m >> 3)] * u[m & 7];      // i=4: x4 (outer) u1
    u[4680 + m] = acc;
  }
  __syncthreads();
  wmma_outer_acc<64, 512>(xl14 + 8, u + 72, u + 4680, tid);  // i=2: x2 (outer) u3
  __syncthreads();
  wmma_outer_acc<512, 64>(xl14 + 72, u + 8, u + 4680, tid);  // i=3: x3 (outer) u2
  __syncthreads();
}

// emit all Lyndon outputs of level K:
// out = x_K[j] + sum_{i=1..K-1} x_i[j >> 3(K-i)] * Q_{K-i}[j & mask]
template <int K>
__device__ __forceinline__ void emit_level(const float* __restrict__ xl14,
                                           const float* __restrict__ u,
                                           const float* __restrict__ x5g,
                                           const float* __restrict__ x6g,
                                           const unsigned* __restrict__ lyn,
                                           float* __restrict__ outb, int tid) {
  for (int pos = OUT_BEG(K) + tid; pos < OUT_END(K); pos += NT) {
    const unsigned j = lyn[pos] & 0xFFFFFu;
    float acc;
    if constexpr (K <= 4)      acc = xl14[LVL_OFF(K) + (int)j];
    else if constexpr (K == 5) acc = x5g[j];
    else                       acc = x6g[j];
#pragma unroll
    for (int i = 1; i < K; ++i) {
      const int s = 3 * (K - i);
      const unsigned pre = j >> s;
      const unsigned suf = j & ((1u << s) - 1u);
      const float xv = (i < 5) ? xl14[LVL_OFF(i) + (int)pre] : x5g[pre];
      acc += xv * u[LVL_OFF(K - i) + (int)suf];
    }
    outb[pos] = acc;
  }
}

// ---------------------------------------------------------------------------
// Kernel 2: one block per batch element. Build Q in LDS, emit Lyndon outputs.
// ---------------------------------------------------------------------------
__global__ void __launch_bounds__(NT) logsig_main(const float* __restrict__ sig,
                                                  const unsigned* __restrict__ lyn,
                                                  float* __restrict__ out) {
  extern __shared__ float smem[];
  float* xl14 = smem;          // x levels 1..4 : 4680 floats
  float* u = smem + 4680;      // U/Q levels 1..5: 37448 floats
  const int tid = (int)threadIdx.x;
  const int b = (int)blockIdx.x;
  const float* xr = sig + (size_t)b * 299592u;
  const float* x5g = xr + 4680;    // level 5 (32768)
  const float* x6g = xr + 37448;   // level 6 (262144)

  for (int t = tid; t < 4680; t += NT) xl14[t] = xr[t];
  // warm L2 with this batch's level-6 row for the scattered Lyndon gather
  for (int t = tid; t < 16384; t += NT)
    __builtin_prefetch((const char*)x6g + (size_t)t * 64, 0, 0);
  __syncthreads();

  // Q = x*(a1 + x*(a2 + x*(a3 + x*(a4 + a5*x)))),  a_n = (-1)^n/(n+1)
  const float a1 = -0.5f, a2 = 1.0f / 3.0f, a3 = -0.25f, a4 = 0.2f, a5 = -1.0f / 6.0f;
  for (int t = tid; t < 8; t += NT) u[t] = a5 * xl14[t];  // U = a4 + a5*x
  __syncthreads();
  // U = a3 + x.U (levels 2..1, descending, in place)
  mul_level<2>(xl14, u, a4, tid); __syncthreads();
  mul_level<1>(xl14, u, a4, tid); __syncthreads();
  // U = a2 + x.U (levels 3..1)
  mul_level<3>(xl14, u, a3, tid); __syncthreads();
  mul_level<2>(xl14, u, a3, tid); __syncthreads();
  mul_level<1>(xl14, u, a3, tid); __syncthreads();
  // U = a1 + x.U (levels 4..1)
  mul_level<4>(xl14, u, a2, tid); __syncthreads();
  mul_level<3>(xl14, u, a2, tid); __syncthreads();
  mul_level<2>(xl14, u, a2, tid); __syncthreads();
  mul_level<1>(xl14, u, a2, tid); __syncthreads();
  // Q = x.U (levels 5..1); level 5 uses the WMMA outer-product path
  mul_level5(xl14, u, x5g, a1, tid);
  mul_level<4>(xl14, u, a1, tid); __syncthreads();
  mul_level<3>(xl14, u, a1, tid); __syncthreads();
  mul_level<2>(xl14, u, a1, tid); __syncthreads();
  mul_level<1>(xl14, u, a1, tid); __syncthreads();

  // Lyndon emission, one specialized loop per level (ranges are static)
  float* outb = out + (size_t)b * 51360u;
  emit_level<1>(xl14, u, x5g, x6g, lyn, outb, tid);
  emit_level<2>(xl14, u, x5g, x6g, lyn, outb, tid);
  emit_level<3>(xl14, u, x5g, x6g, lyn, outb, tid);
  emit_level<4>(xl14, u, x5g, x6g, lyn, outb, tid);
  emit_level<5>(xl14, u, x5g, x6g, lyn, outb, tid);
  emit_level<6>(xl14, u, x5g, x6g, lyn, outb, tid);
}

// ---------------------------------------------------------------------------
extern "C" void kernel_launch(void* const* d_in, const int* in_sizes, int n_in,
                              void* d_out, int out_size, void* d_ws, size_t ws_size,
                              hipStream_t stream) {
  (void)in_sizes; (void)n_in; (void)out_size; (void)ws_size;
  const float* sig = (const float*)d_in[0];
  float* out = (float*)d_out;
  unsigned* lyn = (unsigned*)d_ws;  // 51360 * 4 B = 205,440 B scratch

  lyndon_kernel<<<6, NT, 0, stream>>>(lyn);
  // dynamic LDS: 42128 floats = 168,512 B (fits 320 KB/WGP)
  logsig_main<<<128, NT, 42128 * sizeof(float), stream>>>(sig, lyn, out);
}